// FreeSpacePropagation_84207128805465
// MI455X (gfx1250) — compile-verified
//
#include <hip/hip_runtime.h>
#include <math.h>

typedef __attribute__((ext_vector_type(2))) float v2f;
typedef __attribute__((ext_vector_type(8))) float v8f;

#define NN     192
#define NWAVE  8          // waves per block (wave32)
#define RROWS  207        // 16 + 192 - 1 kernel rows touched per y-tile
#define RITERS 26         // ceil(207 / NWAVE)

__launch_bounds__(256)
__global__ void fsp_wmma_kernel(const float* __restrict__ Fr,
                                const float* __restrict__ Fi,
                                const float* __restrict__ zp,
                                float* __restrict__ out)
{
    // per-wave kernel-row segment: A[r, x0 .. x0+206] as (re, im)
    __shared__ float2 aRow[NWAVE][RROWS + 1];
    // cross-wave reduction buffer: [wave][lane][8 re + 8 im]
    __shared__ float  red[NWAVE][32][16];

    const int tid  = threadIdx.x;
    const int w    = tid >> 5;
    const int lane = tid & 31;
    const int mn   = lane & 15;     // serves as M (A-frag row) and N (B-frag col)
    const int half = lane >> 4;

    const int x0 = blockIdx.x * 16; // x-tile origin (pre-flip coords)
    const int Y0 = blockIdx.y * 16; // y-tile origin (pre-flip coords)

    const float z   = zp[0];
    const float kph = 12.566370614359172f;  // 2*pi / 0.5
    const float C0  = 191.000050f;          // X0 = 0.5*(2N + 1e-4) - 1

    v8f c_re = {};
    v8f c_im = {};

    for (int it = 0; it < RITERS; ++it) {
        const int  r      = Y0 + w + it * NWAVE;
        const bool active = (r < Y0 + RROWS);

        __syncthreads();
        if (active) {
            // Build A[r, x0+u], u in [0,206]: amp = z/R^2, phase = k*R
            const float X    = (float)r - C0;
            const float x2z2 = X * X + z * z;
            for (int u = lane; u < RROWS; u += 32) {
                const float Yc  = (float)(x0 + u) - C0;
                const float R   = sqrtf(x2z2 + Yc * Yc);
                const float amp = z / (R * R);   // cos(atan2(rho,z))/R = z/R^2
                float sp, cp;
                sincosf(kph * R, &sp, &cp);      // accurate range reduction
                aRow[w][u] = make_float2(amp * cp, amp * sp);
            }
        }
        __syncthreads();

        if (active) {
            // field row for this lane's M index: i = r - Y0 - 15 + m
            const int  i     = r - Y0 - 15 + mn;
            const bool iok   = (i >= 0) && (i < NN);
            const int  ibase = (iok ? i : 0) * NN;

            #pragma unroll 4
            for (int s = 0; s < NN / 2; ++s) {
                // A fragment (field, complex interleaved along K):
                // lane(m,half): a = (Fr[i, 2s+half], Fi[i, 2s+half])
                const int col = 2 * s + half;
                const float fx = Fr[ibase + col];
                const float fy = Fi[ibase + col];
                v2f a;
                a.x = iok ? fx : 0.0f;
                a.y = iok ? fy : 0.0f;

                // B fragments (Toeplitz of kernel row):
                // lane(n,half) reads A[r, x0 + n + 2s + half]
                const float2 Av = aRow[w][mn + 2 * s + half];
                v2f b_re, b_im;
                b_re.x =  Av.x;  b_re.y = -Av.y;   // (Ar, -Ai) -> real part
                b_im.x =  Av.y;  b_im.y =  Av.x;   // (Ai,  Ar) -> imag part

                c_re = __builtin_amdgcn_wmma_f32_16x16x4_f32(
                          false, a, false, b_re, (short)0, c_re, false, false);
                c_im = __builtin_amdgcn_wmma_f32_16x16x4_f32(
                          false, a, false, b_im, (short)0, c_im, false, false);
            }
        }
    }

    // ---- deterministic cross-wave reduction + flipped write-out ----
    __syncthreads();
    #pragma unroll
    for (int v = 0; v < 8; ++v) {
        red[w][lane][v]     = c_re[v];
        red[w][lane][v + 8] = c_im[v];
    }
    __syncthreads();

    for (int slot = tid; slot < 512; slot += 256) {
        const int lane_s = slot & 31;
        const int rest   = slot >> 5;   // 0..15
        const int v      = rest & 7;    // C/D VGPR index
        const int part   = rest >> 3;   // 0 = re, 1 = im
        float sum = 0.0f;
        #pragma unroll
        for (int ww = 0; ww < NWAVE; ++ww)
            sum += red[ww][lane_s][part * 8 + v];

        // D layout: n = lane&15, m = v + 8*(lane>>4); pre-flip y = Y0+15-m
        const int m  = v + 8 * (lane_s >> 4);
        const int n  = lane_s & 15;
        const int yf = 176 - Y0 + m;    // 191 - (Y0 + 15 - m)
        const int xf = 191 - x0 - n;    // flipped column
        out[(yf * NN + xf) * 2 + part] = sum;
    }
}

extern "C" void kernel_launch(void* const* d_in, const int* in_sizes, int n_in,
                              void* d_out, int out_size, void* d_ws, size_t ws_size,
                              hipStream_t stream) {
    (void)in_sizes; (void)n_in; (void)d_ws; (void)ws_size; (void)out_size;
    const float* Fr = (const float*)d_in[0];   // input_real [1,1,192,192]
    const float* Fi = (const float*)d_in[1];   // input_imag [1,1,192,192]
    const float* zp = (const float*)d_in[2];   // z (scalar)
    float* out = (float*)d_out;                // complex64 -> interleaved (re,im)

    dim3 grid(NN / 16, NN / 16);               // 12 x 12 output tiles
    dim3 block(256);                           // 8 wave32 waves
    fsp_wmma_kernel<<<grid, block, 0, stream>>>(Fr, Fi, zp, out);
}